// RelaLayer_12627203850784
// MI455X (gfx1250) — compile-verified
//
#include <hip/hip_runtime.h>
#include <hip/hip_bf16.h>

typedef __attribute__((ext_vector_type(2))) float v2f;
typedef __attribute__((ext_vector_type(8))) float v8f;

// ---------------------------------------------------------------------------
// fp32 GEMM via V_WMMA_F32_16X16X4_F32 (exact fp32, matches reference dtype).
// C[M,N] = A[M,K] * B[K,N].  BT=true: B[k][n] = Bmem[n*ldb + k] (B transposed
// in memory).  One wave computes a 16x32 tile: two independent accumulator
// chains sharing the A fragment.
// ---------------------------------------------------------------------------
template <bool BT>
__global__ __launch_bounds__(32) void gemm_wmma(
    const float* __restrict__ A, int lda,
    const float* __restrict__ B, int ldb,
    float* __restrict__ C, int ldc,
    int N, int K)
{
    const int ntn  = N >> 5;                 // number of 32-wide N tiles
    const int tm   = blockIdx.x / ntn;
    const int tn   = blockIdx.x % ntn;
    const int lane = threadIdx.x & 31;
    const int half = lane >> 4;              // 0: lanes 0-15, 1: lanes 16-31
    const int lr   = lane & 15;

    v8f acc0 = {}, acc1 = {};
    const float* Arow = A + (tm * 16 + lr) * lda;
    const float* B0;
    const float* B1;
    if (BT) {                                // B element [k][n] at Bmem[n*ldb+k]
        B0 = B + (tn * 32 + lr) * ldb;
        B1 = B + (tn * 32 + 16 + lr) * ldb;
    } else {                                 // row-major B
        B0 = B + tn * 32 + lr;
        B1 = B0 + 16;
    }

#pragma unroll 4
    for (int k = 0; k < K; k += 4) {
        const int kk = k + 2 * half;         // A frag: v[j] = A[m][kk + j]
        v2f a;  a[0] = Arow[kk];  a[1] = Arow[kk + 1];
        v2f b0, b1;
        if (BT) {
            b0[0] = B0[kk];        b0[1] = B0[kk + 1];
            b1[0] = B1[kk];        b1[1] = B1[kk + 1];
        } else {
            b0[0] = B0[kk * ldb];  b0[1] = B0[(kk + 1) * ldb];
            b1[0] = B1[kk * ldb];  b1[1] = B1[(kk + 1) * ldb];
        }
        acc0 = __builtin_amdgcn_wmma_f32_16x16x4_f32(
            false, a, false, b0, (short)0, acc0, false, false);
        acc1 = __builtin_amdgcn_wmma_f32_16x16x4_f32(
            false, a, false, b1, (short)0, acc1, false, false);
    }

    // C/D layout: VGPR r -> row (r + 8*half), col lr.
    float* C0 = C + (tm * 16 + 8 * half) * ldc + tn * 32 + lr;
#pragma unroll
    for (int r = 0; r < 8; ++r) {
        C0[r * ldc]      = acc0[r];
        C0[r * ldc + 16] = acc1[r];
    }
}

// ave_q[h] = sum_q qry_h[q,h] / Q^2   (faithful to the reference's double division)
__global__ void aveq_kernel(const float* __restrict__ qry_h,
                            float* __restrict__ aveq, int Q, int H)
{
    int h = blockIdx.x * blockDim.x + threadIdx.x;
    if (h >= H) return;
    float s = 0.f;
    for (int q = 0; q < Q; ++q) s += qry_h[q * H + h];
    aveq[h] = s / (float)(Q * Q);
}

// c[i] = (ave_q @ W1[:,2H:].T + b1) @ W2.T + b2
__global__ void c_kernel(const float* __restrict__ aveq,
                         const float* __restrict__ W1,
                         const float* __restrict__ b1,
                         const float* __restrict__ W2,
                         const float* __restrict__ b2,
                         float* __restrict__ cvec, int H)
{
    __shared__ float t[1024];
    int j = threadIdx.x;
    if (j < H) {
        float acc = b1[j];
        const float* W1Cj = W1 + j * 3 * H + 2 * H;    // W1[j, 2H+k]
        for (int k = 0; k < H; ++k) acc += aveq[k] * W1Cj[k];
        t[j] = acc;
    }
    __syncthreads();
    if (j < H) {
        float ci = b2[j];
        const float* W2j = W2 + j * H;
        for (int k = 0; k < H; ++k) ci += t[k] * W2j[k];
        cvec[j] = ci;
    }
}

// us[s,h], vs[s,h]: per-sentence means of uw/vw rows (linearity of the GEMM).
__global__ void usv_kernel(const float* __restrict__ uw,
                           const float* __restrict__ vw,
                           const int* __restrict__ ss,
                           const int* __restrict__ se,
                           float* __restrict__ us, float* __restrict__ vs, int H)
{
    int s = blockIdx.x, h = threadIdx.x;
    int st = ss[s], en = se[s];
    float su = 0.f, sv = 0.f;
    for (int m = st; m <= en; ++m) {
        su += uw[m * H + h];
        sv += vw[m * H + h];
    }
    float inv = 1.f / (float)(en - st + 1);
    us[s * H + h] = su * inv;
    vs[s * H + h] = sv * inv;
}

// P0sum[h] = sum_{a!=b} relu(us[a,h]+vs[b,h]+c[h]); context = P0sum / (S*(S-1))
__global__ void p0_kernel(const float* __restrict__ us,
                          const float* __restrict__ vs,
                          const float* __restrict__ cvec,
                          float* __restrict__ P0, float* __restrict__ ctx,
                          int S, int H)
{
    int h = threadIdx.x;
    float ch = cvec[h];
    float p = 0.f;
    for (int a = 0; a < S; ++a)
        for (int b = 0; b < S; ++b)
            if (a != b)
                p += fmaxf(us[a * H + h] + vs[b * H + h] + ch, 0.f);
    P0[h]  = p;
    ctx[h] = p / (float)(S * (S - 1));
}

// T[m,h] = sum_{a<m} relu(uw[a]+vw[m]+c) + relu(uw[m]+vw[a]+c)
__global__ void pair_T_kernel(const float* __restrict__ uw,
                              const float* __restrict__ vw,
                              const float* __restrict__ cvec,
                              float* __restrict__ T, int H)
{
    int m = blockIdx.x, h = threadIdx.x;
    float ch = cvec[h];
    float um = uw[m * H + h];
    float vm = vw[m * H + h];
    float acc = 0.f;
    for (int a = 0; a < m; ++a) {
        float ua = uw[a * H + h];
        float va = vw[a * H + h];
        acc += fmaxf(ua + vm + ch, 0.f);
        acc += fmaxf(um + va + ch, 0.f);
    }
    T[m * H + h] = acc;
}

// Tsent[s,h] = sum of T rows inside sentence s
__global__ void tsent_kernel(const float* __restrict__ T,
                             const int* __restrict__ ss,
                             const int* __restrict__ se,
                             float* __restrict__ Tsent, int H)
{
    int s = blockIdx.x, h = threadIdx.x;
    float acc = 0.f;
    for (int m = ss[s]; m <= se[s]; ++m) acc += T[m * H + h];
    Tsent[s * H + h] = acc;
}

// scv[i,h] = ((P0sum + Sall[i]) / cnt[i] + context) * 0.5
// Sall[i] = sum_{s<=i} (i-s+1) * Tsent[s];  cnt[i] = S(S-1) + cumsum N(N-1)
__global__ void scv_kernel(const float* __restrict__ Tsent,
                           const float* __restrict__ P0,
                           const float* __restrict__ ctx,
                           const int* __restrict__ se,
                           float* __restrict__ scv, int S, int H)
{
    int i = blockIdx.x, h = threadIdx.x;
    float sall = 0.f;
    for (int s = 0; s <= i; ++s)
        sall += (float)(i - s + 1) * Tsent[s * H + h];
    float cnt = (float)(S * (S - 1));
    for (int j = 0; j <= i; ++j) {
        float N = (float)(se[j] + 1);
        cnt += N * (N - 1.f);
    }
    float sv = (P0[h] + sall) / cnt;
    scv[i * H + h] = (sv + ctx[h]) * 0.5f;
}

// out[m,h] = covered ? (doc_h + scv[seg(m)]) * 0.5 : doc
__global__ void out_kernel(const float* __restrict__ doc_h,
                           const float* __restrict__ doc,
                           const float* __restrict__ scv,
                           const int* __restrict__ se,
                           float* __restrict__ out, int D, int H, int S)
{
    int idx = blockIdx.x * blockDim.x + threadIdx.x;
    if (idx >= D * H) return;
    int m = idx / H;
    int h = idx - m * H;
    if (m <= se[S - 1]) {
        int s = 0;                            // searchsorted(sent_end, m)
        while (s < S - 1 && se[s] < m) ++s;
        out[idx] = (doc_h[idx] + scv[s * H + h]) * 0.5f;
    } else {
        out[idx] = doc[idx];
    }
}

extern "C" void kernel_launch(void* const* d_in, const int* in_sizes, int n_in,
                              void* d_out, int out_size, void* d_ws, size_t ws_size,
                              hipStream_t stream)
{
    const float* doc_h = (const float*)d_in[0];
    const float* qry_h = (const float*)d_in[1];
    const float* doc   = (const float*)d_in[2];
    /* d_in[3] (qry) unused by forward */
    const int*   ss    = (const int*)d_in[4];
    const int*   se    = (const int*)d_in[5];
    const float* W1    = (const float*)d_in[6];
    const float* b1    = (const float*)d_in[7];
    const float* W2    = (const float*)d_in[8];
    const float* b2    = (const float*)d_in[9];

    const int H = in_sizes[7];            // b1 has H elements
    const int S = in_sizes[4];
    const int D = in_sizes[0] / H;        // B == 1
    const int Q = in_sizes[1] / H;

    float* ws = (float*)d_ws;
    size_t off = 0;
    float* aveq  = ws + off; off += (size_t)H;
    float* UA    = ws + off; off += (size_t)H * H;
    float* VB    = ws + off; off += (size_t)H * H;
    float* cvec  = ws + off; off += (size_t)H;
    float* uw    = ws + off; off += (size_t)D * H;
    float* vw    = ws + off; off += (size_t)D * H;
    float* P0    = ws + off; off += (size_t)H;
    float* ctx   = ws + off; off += (size_t)H;
    float* T     = ws + off; off += (size_t)D * H;
    float* us    = ws + off; off += (size_t)S * H;
    float* vs    = ws + off; off += (size_t)S * H;
    float* Tsent = ws + off; off += (size_t)S * H;
    float* scv   = ws + off; off += (size_t)S * H;
    (void)ws_size; (void)n_in; (void)out_size;

    dim3 b32(32), b256(256);

    aveq_kernel<<<(H + 255) / 256, b256, 0, stream>>>(qry_h, aveq, Q, H);

    // UA = W2 @ W1[:, 0:H] ; VB = W2 @ W1[:, H:2H]   (row-major B, ldb = 3H)
    dim3 gHH((H / 16) * (H / 32));
    gemm_wmma<false><<<gHH, b32, 0, stream>>>(W2, H, W1,     3 * H, UA, H, H, H);
    gemm_wmma<false><<<gHH, b32, 0, stream>>>(W2, H, W1 + H, 3 * H, VB, H, H, H);

    c_kernel<<<1, H, 0, stream>>>(aveq, W1, b1, W2, b2, cvec, H);

    // uw = doc_h @ UA.T ; vw = doc_h @ VB.T   (B[k][n] = UA[n*H + k])
    dim3 gDH((D / 16) * (H / 32));
    gemm_wmma<true><<<gDH, b32, 0, stream>>>(doc_h, H, UA, H, uw, H, H, H);
    gemm_wmma<true><<<gDH, b32, 0, stream>>>(doc_h, H, VB, H, vw, H, H, H);

    usv_kernel<<<S, H, 0, stream>>>(uw, vw, ss, se, us, vs, H);
    p0_kernel<<<1, H, 0, stream>>>(us, vs, cvec, P0, ctx, S, H);
    pair_T_kernel<<<D, H, 0, stream>>>(uw, vw, cvec, T, H);
    tsent_kernel<<<S, H, 0, stream>>>(T, ss, se, Tsent, H);
    scv_kernel<<<S, H, 0, stream>>>(Tsent, P0, ctx, se, scv, S, H);

    int total = D * H;
    out_kernel<<<(total + 255) / 256, b256, 0, stream>>>(
        doc_h, doc, scv, se, (float*)d_out, D, H, S);
}